// IntegralConvEmbedding_23751169147523
// MI455X (gfx1250) — compile-verified
//
#include <hip/hip_runtime.h>
#include <math.h>

// ---------------------------------------------------------------------------
// IntegralConvEmbedding forward for MI455X (gfx1250, wave32, WMMA).
//
// Input flattening assumption (JAX pytree: dict keys sorted alphabetically):
//   0: q (128*16384)   1: y (128*16384*2)
//   per z-block (base 2 for nz=256, base 22 for nz=512), 20 leaves:
//     +0 conv0.W +1 conv0.b +2 conv1.W +3 conv1.b +4 conv2.W +5 conv2.b
//     +6 head.W  +7 head.b
//     +8 fc.fc1.W +9 fc.fc1.b +10 fc.fc2.W +11 fc.fc2.b +12 fc.ln_b +13 fc.ln_g
//     +14 k.fc1.W +15 k.fc1.b +16 k.fc2.W +17 k.fc2.b +18 k.ln_b +19 k.ln_g
//   res (base 42): 4 x {lin1.W,lin1.b,lin2.W,lin2.b,ln_b,ln_g} then
//     66 first.W 67 first.b 68 last.W 69 last.b
// ---------------------------------------------------------------------------

typedef __attribute__((ext_vector_type(2))) float v2f;
typedef __attribute__((ext_vector_type(8))) float v8f;

#define BATCHN   128
#define NPTSN    16384
#define PK_CHUNK 2048        // points per workgroup in the point kernel

__device__ __forceinline__ v8f wmma4(v2f a, v2f b, v8f c) {
  // V_WMMA_F32_16X16X4_F32 : D(16x16 f32) = A(16x4) * B(4x16) + C
  return __builtin_amdgcn_wmma_f32_16x16x4_f32(false, a, false, b, (short)0, c,
                                               false, false);
}

__device__ __forceinline__ float gelu_exact(float x) {
  return 0.5f * x * (1.0f + erff(x * 0.70710678118654752440f));
}

// XOR-butterfly add across each 16-lane row using v_permlane16_b32 (pure VALU;
// avoids the ds_bpermute + s_wait_dscnt chains that __shfl_xor lowers to).
// Nibble i of {selHi,selLo} = source lane for dst lane i, i.e. i^m.
__device__ __forceinline__ float permxor16(float x, unsigned selLo, unsigned selHi) {
  int v = __float_as_int(x);
  int p = __builtin_amdgcn_permlane16(v, v, (int)selLo, (int)selHi, false, false);
  return __int_as_float(p);
}
__device__ __forceinline__ float half16_sum(float x) {
  x += permxor16(x, 0x67452301u, 0xEFCDAB89u);  // xor 1
  x += permxor16(x, 0x54761032u, 0xDCFE98BAu);  // xor 2
  x += permxor16(x, 0x32107654u, 0xBA98FEDCu);  // xor 4
  x += permxor16(x, 0xFEDCBA98u, 0x76543210u);  // xor 8
  return x;
}

// ---------------------------------------------------------------------------
// Fused per-point kernel: bucketize -> fc1 (WMMA 16x16x4, K=4 exact) -> LN ->
// exact GELU -> fc2 (cross-lane dot) -> *y -> LDS segment bins -> global flush.
// One workgroup = 256 threads (8 waves) handles one (batch, 2048-point chunk).
// ---------------------------------------------------------------------------
template <int NZ>
__device__ __forceinline__ void point_zblock(
    float qv, float y0, float y1, int half, int l,
    v2f B0, v2f B1, float bias0, float bias1,
    float g0, float g1, float be0, float be1,
    float w20A, float w20B, float w21A, float w21B, float b20, float b21,
    float (*st)[2], float* bins) {
  const float scale = (float)(NZ - 1);
  const float invscale = 1.0f / (float)(NZ - 1);
  int idx = (int)ceilf(qv * scale - 0.5f);
  idx = idx < 0 ? 0 : (idx > NZ - 1 ? NZ - 1 : idx);
  float zv = (float)idx * invscale;

  // A tile (16 points x K=4): kin = {q, zv, y0, y1}.
  // f32 A layout: lane half 0 holds K=0,1; half 1 holds K=2,3 (M = lane&15).
  v2f a;
  a.x = half ? y0 : qv;
  a.y = half ? y1 : zv;

  v8f h0, h1;
#pragma unroll
  for (int i = 0; i < 8; ++i) { h0[i] = bias0; h1[i] = bias1; }
  h0 = wmma4(a, B0, h0);   // hidden cols 0..15
  h1 = wmma4(a, B1, h1);   // hidden cols 16..31

  // Row M (a point) lives in VGPR r = M%8 across one 16-lane half.
#pragma unroll
  for (int r = 0; r < 8; ++r) {
    float mean = half16_sum(h0[r] + h1[r]) * (1.0f / 32.0f);
    float d0 = h0[r] - mean, d1 = h1[r] - mean;
    float var = half16_sum(d0 * d0 + d1 * d1) * (1.0f / 32.0f);
    float inv = rsqrtf(var + 1e-5f);
    float x0 = gelu_exact(d0 * inv * g0 + be0);
    float x1 = gelu_exact(d1 * inv * g1 + be1);
    // fc2: out_d[M] = sum_N x[M][N] * W2[d][N]
    float o0 = half16_sum(x0 * w20A + x1 * w20B);
    float o1 = half16_sum(x0 * w21A + x1 * w21B);
    if (l == 0) {                 // lane 0 of each half stages its row
      int M = r + half * 8;
      st[M][0] = o0 + b20;
      st[M][1] = o1 + b21;
    }
  }
  __syncthreads();                // uniform control flow; orders LDS stage
  if (half == 0) {                // lane l owns point l of this wave's tile
    float o0 = st[l][0] * y0;
    float o1 = st[l][1] * y1;
    atomicAdd(&bins[idx * 3 + 0], o0);
    atomicAdd(&bins[idx * 3 + 1], o1);
    atomicAdd(&bins[idx * 3 + 2], 1.0f);
  }
  __syncthreads();
}

__global__ __launch_bounds__(256) void point_kernel(
    const float* __restrict__ q, const float* __restrict__ y,
    const float* w1a, const float* b1a, const float* w2a, const float* b2a,
    const float* lnba, const float* lnga,
    const float* w1b, const float* b1b, const float* w2b, const float* b2b,
    const float* lnbb, const float* lngb,
    float* sums256, float* counts256, float* sums512, float* counts512) {
  __shared__ float bins0[256 * 3];
  __shared__ float bins1[512 * 3];
  __shared__ float stage[8][16][2];

  int tid = threadIdx.x;
  int wave = tid >> 5;
  int lane = tid & 31;
  int half = lane >> 4;
  int l = lane & 15;

  for (int i = tid; i < 256 * 3; i += 256) bins0[i] = 0.f;
  for (int i = tid; i < 512 * 3; i += 256) bins1[i] = 0.f;
  __syncthreads();

  int wg = blockIdx.x;
  int b = wg / (NPTSN / PK_CHUNK);
  int chunk = wg % (NPTSN / PK_CHUNK);
  int base = b * NPTSN + chunk * PK_CHUNK;

  // Preload B tiles (W1^T) for both z-blocks: B[k][n] = W1[n][k].
  v2f B0a, B1a, B0b, B1b;
  B0a.x = w1a[l * 4 + half * 2 + 0];        B0a.y = w1a[l * 4 + half * 2 + 1];
  B1a.x = w1a[(l + 16) * 4 + half * 2 + 0]; B1a.y = w1a[(l + 16) * 4 + half * 2 + 1];
  B0b.x = w1b[l * 4 + half * 2 + 0];        B0b.y = w1b[l * 4 + half * 2 + 1];
  B1b.x = w1b[(l + 16) * 4 + half * 2 + 0]; B1b.y = w1b[(l + 16) * 4 + half * 2 + 1];
  float bias0a = b1a[l], bias1a = b1a[l + 16];
  float bias0b = b1b[l], bias1b = b1b[l + 16];
  float g0a = lnga[l], g1a = lnga[l + 16], be0a = lnba[l], be1a = lnba[l + 16];
  float g0b = lngb[l], g1b = lngb[l + 16], be0b = lnbb[l], be1b = lnbb[l + 16];
  float w20Aa = w2a[l], w20Ba = w2a[l + 16];
  float w21Aa = w2a[32 + l], w21Ba = w2a[32 + l + 16];
  float b20a = b2a[0], b21a = b2a[1];
  float w20Ab = w2b[l], w20Bb = w2b[l + 16];
  float w21Ab = w2b[32 + l], w21Bb = w2b[32 + l + 16];
  float b20b = b2b[0], b21b = b2b[1];

  for (int round = 0; round < PK_CHUNK / 128; ++round) {
    int p = base + round * 128 + wave * 16 + l;   // lanes L and L+16 share a point
    float qv = q[p];
    float yy0 = y[2 * p], yy1 = y[2 * p + 1];
    point_zblock<256>(qv, yy0, yy1, half, l, B0a, B1a, bias0a, bias1a,
                      g0a, g1a, be0a, be1a, w20Aa, w20Ba, w21Aa, w21Ba,
                      b20a, b21a, stage[wave], bins0);
    point_zblock<512>(qv, yy0, yy1, half, l, B0b, B1b, bias0b, bias1b,
                      g0b, g1b, be0b, be1b, w20Ab, w20Bb, w21Ab, w21Bb,
                      b20b, b21b, stage[wave], bins1);
  }
  __syncthreads();
  for (int i = tid; i < 256 * 3; i += 256) {
    int z = i / 3, r = i % 3;
    float v = bins0[i];
    if (r < 2) atomicAdd(&sums256[(b * 256 + z) * 2 + r], v);
    else       atomicAdd(&counts256[b * 256 + z], v);
  }
  for (int i = tid; i < 512 * 3; i += 256) {
    int z = i / 3, r = i % 3;
    float v = bins1[i];
    if (r < 2) atomicAdd(&sums512[(b * 512 + z) * 2 + r], v);
    else       atomicAdd(&counts512[b * 512 + z], v);
  }
}

// sums/counts -> k laid out (b, in_dim, nz) directly inside the feat buffer.
__global__ void finalize_means(const float* __restrict__ sums,
                               const float* __restrict__ counts,
                               float* __restrict__ kout, int ldk, int nz) {
  int i = blockIdx.x * blockDim.x + threadIdx.x;
  if (i >= BATCHN * nz) return;
  int b = i / nz, z = i % nz;
  float c = fmaxf(counts[i], 1.0f);
  kout[b * ldk + z]      = sums[i * 2 + 0] / c;
  kout[b * ldk + nz + z] = sums[i * 2 + 1] / c;
}

// Conv1d k=3 s=2 p=1 + LeakyReLU(0.01). in batch-stride parameterized so the
// first conv can read k straight out of the feat buffer.
__global__ void conv1d_s2_lrelu(const float* __restrict__ in, int inBatchStride,
                                const float* __restrict__ W,
                                const float* __restrict__ bias,
                                float* __restrict__ out,
                                int Cin, int L, int Cout, int Lout) {
  int i = blockIdx.x * blockDim.x + threadIdx.x;
  if (i >= BATCHN * Cout * Lout) return;
  int lpos = i % Lout;
  int t = i / Lout;
  int co = t % Cout;
  int b = t / Cout;
  float acc = bias[co];
  const float* wrow = W + co * Cin * 3;
  const float* inb = in + b * inBatchStride;
  int il0 = 2 * lpos - 1;
  for (int ci = 0; ci < Cin; ++ci) {
    const float* x = inb + ci * L;
    const float* w = wrow + ci * 3;
    if (il0 >= 0) acc += w[0] * x[il0];
    acc += w[1] * x[il0 + 1];
    if (il0 + 2 < L) acc += w[2] * x[il0 + 2];
  }
  out[i] = acc >= 0.f ? acc : 0.01f * acc;
}

// adaptive avg-pool to 8 + flatten (c-major): pooled[b][c*8+p]
__global__ void pool_kernel(const float* __restrict__ in,
                            float* __restrict__ pooled, int L, float invgl) {
  int i = blockIdx.x * blockDim.x + threadIdx.x;
  if (i >= BATCHN * 1024) return;
  int p = i % 8;
  int t = i / 8;
  int c = t % 128;
  int b = t / 128;
  int gl = L / 8;
  const float* x = in + (b * 128 + c) * L + p * gl;
  float s = 0.f;
  for (int j = 0; j < gl; ++j) s += x[j];
  pooled[i] = s * invgl;
}

// Generic WMMA f32 16x16 GEMM tile kernel: out[m][ocol+n] =
//   act(X(MxK,ldx) @ W(NxK,ldw)^T + bias) (+ residual). One wave per tile.
// Two independent WMMA accumulator chains (K % 8 == 0) + b64 vector loads.
__global__ __launch_bounds__(32) void gemm16_kernel(
    const float* __restrict__ X, int ldx, const float* __restrict__ W, int ldw,
    const float* __restrict__ bias, const float* res, int ldres,
    float* out, int ldo, int ocol, int M, int N, int K, int act) {
  int lane = threadIdx.x;
  int half = lane >> 4;
  int l = lane & 15;
  int n0 = blockIdx.x * 16;
  int m0 = blockIdx.y * 16;

  float bv = bias ? bias[n0 + l] : 0.0f;
  v8f acc0, acc1;
#pragma unroll
  for (int i = 0; i < 8; ++i) { acc0[i] = bv; acc1[i] = 0.0f; }

  // even-float offsets -> 8-byte aligned -> legal b64 vector loads
  const float* xrow = X + (m0 + l) * ldx + half * 2;
  const float* wrow = W + (n0 + l) * ldw + half * 2;
  for (int k = 0; k < K; k += 8) {
    v2f a0 = *(const v2f*)(xrow + k);
    v2f b0 = *(const v2f*)(wrow + k);
    acc0 = wmma4(a0, b0, acc0);
    v2f a1 = *(const v2f*)(xrow + k + 4);
    v2f b1 = *(const v2f*)(wrow + k + 4);
    acc1 = wmma4(a1, b1, acc1);
  }
  int mrow = m0 + half * 8;
#pragma unroll
  for (int r = 0; r < 8; ++r) {
    float v = acc0[r] + acc1[r];
    if (act == 1) v = fmaxf(v, 0.0f);
    int m = mrow + r;
    if (res) v += res[m * ldres + n0 + l];
    out[m * ldo + ocol + n0 + l] = v;
  }
}

// Row LayerNorm (width<=512 here), optional exact GELU.
__global__ void layernorm_kernel(const float* __restrict__ x,
                                 const float* __restrict__ g,
                                 const float* __restrict__ bb,
                                 float* __restrict__ out, int width, int dogelu) {
  __shared__ float red[256];
  int row = blockIdx.x, tid = threadIdx.x;
  const float* xr = x + row * width;
  float invw = 1.0f / (float)width;
  float s = 0.f;
  for (int j = tid; j < width; j += 256) s += xr[j];
  red[tid] = s; __syncthreads();
  for (int o = 128; o > 0; o >>= 1) { if (tid < o) red[tid] += red[tid + o]; __syncthreads(); }
  float mean = red[0] * invw;
  __syncthreads();
  float v = 0.f;
  for (int j = tid; j < width; j += 256) { float d = xr[j] - mean; v += d * d; }
  red[tid] = v; __syncthreads();
  for (int o = 128; o > 0; o >>= 1) { if (tid < o) red[tid] += red[tid + o]; __syncthreads(); }
  float inv = rsqrtf(red[0] * invw + 1e-5f);
  for (int j = tid; j < width; j += 256) {
    float u = (xr[j] - mean) * inv * g[j] + bb[j];
    if (dogelu) u = gelu_exact(u);
    out[row * width + j] = u;
  }
}

// ---------------------------------------------------------------------------
extern "C" void kernel_launch(void* const* d_in, const int* in_sizes, int n_in,
                              void* d_out, int out_size, void* d_ws, size_t ws_size,
                              hipStream_t stream) {
  (void)in_sizes; (void)n_in; (void)out_size; (void)ws_size;
  const float* P[70];
  for (int i = 0; i < 70 && i < n_in; ++i) P[i] = (const float*)d_in[i];
  const float* q = P[0];
  const float* y = P[1];
  enum { C0W = 0, C0B, C1W, C1B, C2W, C2B, HW, HB,
         F1W, F1B, F2W, F2B, FLNB, FLNG, K1W, K1B, K2W, K2B, KLNB, KLNG };
  const int BI[2] = {2, 22};
  const int RBASE = 42;

  float* ws = (float*)d_ws;
  float* sums256  = ws;                    // 65536
  float* counts256 = sums256 + 65536;      // 32768
  float* sums512  = counts256 + 32768;     // 131072
  float* counts512 = sums512 + 131072;     // 65536
  float* feat256  = counts512 + 65536;     // 128*768
  float* feat512  = feat256 + 128 * 768;   // 128*1280
  float* cbufA    = feat512 + 128 * 1280;  // 1048576
  float* cbufB    = cbufA + 1048576;       // 1048576
  float* pooled   = cbufB + 1048576;       // 128*1024
  float* hbuf     = pooled + 128 * 1024;   // 128*512
  float* hln      = hbuf + 128 * 512;
  float* xcat     = hln + 128 * 512;
  float* xres     = xcat + 128 * 512;
  float* h2       = xres + 128 * 512;

  hipMemsetAsync(sums256, 0, (65536 + 32768 + 131072 + 65536) * sizeof(float), stream);

  point_kernel<<<BATCHN * (NPTSN / PK_CHUNK), 256, 0, stream>>>(
      q, y,
      P[BI[0] + K1W], P[BI[0] + K1B], P[BI[0] + K2W], P[BI[0] + K2B],
      P[BI[0] + KLNB], P[BI[0] + KLNG],
      P[BI[1] + K1W], P[BI[1] + K1B], P[BI[1] + K2W], P[BI[1] + K2B],
      P[BI[1] + KLNB], P[BI[1] + KLNG],
      sums256, counts256, sums512, counts512);

  finalize_means<<<(BATCHN * 256 + 255) / 256, 256, 0, stream>>>(
      sums256, counts256, feat256, 768, 256);
  finalize_means<<<(BATCHN * 512 + 255) / 256, 256, 0, stream>>>(
      sums512, counts512, feat512, 1280, 512);

  auto gemm = [&](const float* X, int ldx, const float* W, int ldw,
                  const float* bias, const float* res, int ldres, float* out,
                  int ldo, int ocol, int M, int N, int K, int act) {
    dim3 g(N / 16, M / 16);
    gemm16_kernel<<<g, 32, 0, stream>>>(X, ldx, W, ldw, bias, res, ldres,
                                        out, ldo, ocol, M, N, K, act);
  };

  for (int j = 0; j < 2; ++j) {
    int base = BI[j];
    int nz = j ? 512 : 256;
    int fd = 2 * nz + 256;
    float* feat = j ? feat512 : feat256;
    int L = nz, L1 = nz / 2, L2 = nz / 4, L3 = nz / 8;

    conv1d_s2_lrelu<<<(BATCHN * 32 * L1 + 255) / 256, 256, 0, stream>>>(
        feat, fd, P[base + C0W], P[base + C0B], cbufA, 2, L, 32, L1);
    conv1d_s2_lrelu<<<(BATCHN * 64 * L2 + 255) / 256, 256, 0, stream>>>(
        cbufA, 32 * L1, P[base + C1W], P[base + C1B], cbufB, 32, L1, 64, L2);
    conv1d_s2_lrelu<<<(BATCHN * 128 * L3 + 255) / 256, 256, 0, stream>>>(
        cbufB, 64 * L2, P[base + C2W], P[base + C2B], cbufA, 64, L2, 128, L3);
    pool_kernel<<<(BATCHN * 1024 + 255) / 256, 256, 0, stream>>>(
        cbufA, pooled, L3, 8.0f / (float)L3);

    // conv head lands directly in feat cols [2*nz, 2*nz+256)
    gemm(pooled, 1024, P[base + HW], 1024, P[base + HB], nullptr, 0,
         feat, fd, 2 * nz, BATCHN, 256, 1024, 0);
    // fcblock: fc1 -> LN+GELU -> fc2 (fc2 lands in concat buffer at col 256*j)
    gemm(feat, fd, P[base + F1W], fd, P[base + F1B], nullptr, 0,
         hbuf, 512, 0, BATCHN, 512, fd, 0);
    layernorm_kernel<<<BATCHN, 256, 0, stream>>>(hbuf, P[base + FLNG],
                                                 P[base + FLNB], hln, 512, 1);
    gemm(hln, 512, P[base + F2W], 512, P[base + F2B], nullptr, 0,
         xcat, 512, 256 * j, BATCHN, 256, 512, 0);
  }

  // residual MLP
  gemm(xcat, 512, P[66], 512, P[67], nullptr, 0, xres, 512, 0, BATCHN, 512, 512, 0);
  for (int r = 0; r < 4; ++r) {
    int rb = RBASE + 6 * r;  // lin1.W, lin1.b, lin2.W, lin2.b, ln_b, ln_g
    layernorm_kernel<<<BATCHN, 256, 0, stream>>>(xres, P[rb + 5], P[rb + 4],
                                                 hln, 512, 0);
    gemm(hln, 512, P[rb + 0], 512, P[rb + 1], nullptr, 0, h2, 512, 0,
         BATCHN, 512, 512, 1 /*relu*/);
    gemm(h2, 512, P[rb + 2], 512, P[rb + 3], xres, 512, xres, 512, 0,
         BATCHN, 512, 512, 0);
  }
  gemm(xres, 512, P[68], 512, P[69], nullptr, 0, (float*)d_out, 256, 0,
       BATCHN, 256, 512, 0);
}